// DynamicWeightedMSELoss_22454089023779
// MI455X (gfx1250) — compile-verified
//
#include <hip/hip_runtime.h>
#include <hip/hip_bf16.h>

#define TOTAL_ELEMS 20971520u          // 4194304 * 5
#define NCHUNK      5242880u           // TOTAL_ELEMS / 4 (float4 chunks)
#define NBINS       201
#define NCH         5
#define PADBINS     203                // [0]=1.0 sentinel, [1..201]=weights, [202]=1.0 sentinel
#define LUTCH       8                  // duplicated channels so (c0+e) needs no mod
#define LUT_SIZE    (LUTCH * PADBINS)  // 1624 floats
#define MAIN_BLOCKS 2560
#define THREADS     256
#define NWAVES      (THREADS / 32)
#define ITERS       8                  // NCHUNK / (MAIN_BLOCKS*THREADS) exactly

static_assert(MAIN_BLOCKS * THREADS * ITERS == NCHUNK, "grid must tile chunks exactly");

typedef __attribute__((ext_vector_type(2))) float v2f;
typedef __attribute__((ext_vector_type(8))) float v8f;

// Wave32 all-lanes sum via the CDNA5 matrix pipe: lane partials form a 16x4
// f32 A-matrix; B = all-ones makes D[m][n] = rowsum(m). Lanes 0-15 hold
// rowsums 0..7 in d[0..7], lanes 16-31 hold rowsums 8..15; one xor-16
// shuffle completes the sum in every lane.
__device__ __forceinline__ float wave_sum_all(float p) {
#if __has_builtin(__builtin_amdgcn_wmma_f32_16x16x4_f32)
    v2f a = {p, 0.0f};
    v2f b = {1.0f, 1.0f};
    v8f c = {};
    v8f d = __builtin_amdgcn_wmma_f32_16x16x4_f32(
        /*neg_a=*/false, a, /*neg_b=*/false, b,
        /*c_mod=*/(short)0, c, /*reuse_a=*/false, /*reuse_b=*/false);
    float s = 0.0f;
#pragma unroll
    for (int i = 0; i < 8; ++i) s += d[i];
    s += __shfl_xor(s, 16, 32);
    return s;
#else
    float s = p;
#pragma unroll
    for (int off = 16; off > 0; off >>= 1) s += __shfl_xor(s, off, 32);
    return s;
#endif
}

// Pass 1: padded+duplicated weight LUT.
// wtab[ch*203 + 0] = wtab[ch*203 + 202] = 1.0 (out-of-range sentinels)
// wtab[ch*203 + 1 + j] = 1 - counts[ch%5][j]/total[ch%5]
// Totals are exact in i32 (max ~2.01e7 < 2^31), converted once to f32.
__global__ void dwmse_build_weights(const int* __restrict__ counts,
                                    float* __restrict__ wtab) {
    __shared__ float tot[NCH];
    if (threadIdx.x < NCH) {
        int t = 0;
        const int* row = counts + threadIdx.x * NBINS;
        for (int i = 0; i < NBINS; ++i) t += row[i];
        tot[threadIdx.x] = (float)t;
    }
    __syncthreads();
    for (int j = threadIdx.x; j < LUT_SIZE; j += blockDim.x) {
        int ch  = j / PADBINS;
        int bin = j - ch * PADBINS;          // 0..202
        int c   = ch - ((ch >= NCH) ? NCH : 0);  // ch % 5 (ch <= 7)
        float w = 1.0f;
        if (bin >= 1 && bin <= NBINS) {
            w = 1.0f - (float)counts[c * NBINS + (bin - 1)] / tot[c];
        }
        wtab[j] = w;
    }
}

// Pass 2: HBM-bandwidth-bound stream. Branch-free weight lookup:
// bin = rintf(x*10)+100 (round-half-even == jnp.round; bin centers are the
// exact same fp32 values k/10, so in-range => exact match in the reference).
// idx = clamp(bin+1, 0, 202) lands out-of-range bins on the 1.0 sentinels.
__global__ void dwmse_main(const float* __restrict__ in,
                           const float* __restrict__ tgt,
                           const float* __restrict__ wtab,
                           float* __restrict__ partials) {
    __shared__ float lw[LUT_SIZE];
    __shared__ float wsum[NWAVES];
    for (int i = threadIdx.x; i < LUT_SIZE; i += blockDim.x) lw[i] = wtab[i];
    __syncthreads();

    const float4* in4 = (const float4*)in;
    const float4* tg4 = (const float4*)tgt;
    const unsigned stride = MAIN_BLOCKS * THREADS;
    const unsigned tid = blockIdx.x * blockDim.x + threadIdx.x;

    float acc = 0.0f;
#pragma unroll
    for (unsigned k = 0; k < ITERS; ++k) {
        const unsigned i = tid + k * stride;
        // keep the load pipe primed one grid-stride ahead
        // (global_prefetch_b8; speculative, OOB prefetches dropped by HW)
        __builtin_prefetch(&in4[i + stride], 0, 0);
        __builtin_prefetch(&tg4[i + stride], 0, 0);

        float4 x = in4[i];   // global_load_b128
        float4 t = tg4[i];   // global_load_b128
        float xa[4] = {x.x, x.y, x.z, x.w};
        float ta[4] = {t.x, t.y, t.z, t.w};
        const unsigned c0 = (i * 4u) % 5u;   // channel of first element; c0+3 <= 7 < LUTCH
#pragma unroll
        for (int e = 0; e < 4; ++e) {
            float d = xa[e] - ta[e];
            int bin = (int)__builtin_rintf(xa[e] * 10.0f) + 100;
            int idx = bin + 1;
            idx = idx < 0 ? 0 : (idx > 202 ? 202 : idx);      // v_med3_i32
            float w = lw[(c0 + e) * PADBINS + idx];           // unconditional ds_load_b32
            acc = fmaf(w * d, d, acc);
        }
    }

    float s = wave_sum_all(acc);          // v_wmma_f32_16x16x4_f32 reduction
    if ((threadIdx.x & 31) == 0) wsum[threadIdx.x >> 5] = s;
    __syncthreads();
    if (threadIdx.x == 0) {
        float t = 0.0f;
#pragma unroll
        for (int w = 0; w < NWAVES; ++w) t += wsum[w];
        partials[blockIdx.x] = t;
    }
}

// Pass 3: deterministic fixed-order reduction of block partials -> mean.
__global__ void dwmse_final(const float* __restrict__ partials, int n,
                            float* __restrict__ out) {
    __shared__ float wsum[NWAVES];
    float acc = 0.0f;
    for (int i = threadIdx.x; i < n; i += blockDim.x) acc += partials[i];
    float s = wave_sum_all(acc);
    if ((threadIdx.x & 31) == 0) wsum[threadIdx.x >> 5] = s;
    __syncthreads();
    if (threadIdx.x == 0) {
        float t = 0.0f;
#pragma unroll
        for (int w = 0; w < NWAVES; ++w) t += wsum[w];
        out[0] = t * (1.0f / (float)TOTAL_ELEMS);
    }
}

extern "C" void kernel_launch(void* const* d_in, const int* in_sizes, int n_in,
                              void* d_out, int out_size, void* d_ws, size_t ws_size,
                              hipStream_t stream) {
    const float* in     = (const float*)d_in[0];  // input  [B,5] f32
    const float* tgt    = (const float*)d_in[1];  // target [B,5] f32
    // d_in[2] = steps [5,201] f32 (bin centers are analytically (k-100)/10 in
    // fp32 — identical bits to the rounded inputs, so no device search needed)
    const int*   counts = (const int*)d_in[3];    // counts [5,201] i32

    float* wtab     = (float*)d_ws;                      // LUT_SIZE floats (6.5 KB)
    float* partials = (float*)((char*)d_ws + 8192);      // MAIN_BLOCKS floats

    dwmse_build_weights<<<1, THREADS, 0, stream>>>(counts, wtab);
    dwmse_main<<<MAIN_BLOCKS, THREADS, 0, stream>>>(in, tgt, wtab, partials);
    dwmse_final<<<1, THREADS, 0, stream>>>(partials, MAIN_BLOCKS, (float*)d_out);
}